// SparseVoxelGrid_17514876634255
// MI455X (gfx1250) — compile-verified
//
#include <hip/hip_runtime.h>
#include <math.h>

typedef __attribute__((ext_vector_type(2))) float v2f;
typedef __attribute__((ext_vector_type(8))) float v8f;

#define EPS_F 1e-8f
// exp(-0.5*x) == exp2(-0.5*log2(e)*x)
#define NEG_HALF_LOG2E (-0.72134752044448170367996234050095f)

__device__ __forceinline__ float fast_w(float d2) {
    const float dist = __builtin_amdgcn_sqrtf(fmaxf(d2, 0.0f)); // v_sqrt_f32
    return __builtin_amdgcn_exp2f(NEG_HALF_LOG2E * dist);       // v_exp_f32
}

// ---------------- zero workspace ----------------
__global__ void svg_zero_ws(float* __restrict__ ws, int n) {
    int i = blockIdx.x * blockDim.x + threadIdx.x;
    if (i < n) ws[i] = 0.0f;
}

// ---------------- pass 1: row sums of exp(-dist/2) via WMMA ----------------
// One wave (32 threads) per block. Block handles 16 query rows x mPer voxels.
__global__ void svg_rowsum(const float* __restrict__ q,
                           const float* __restrict__ c,
                           float* __restrict__ rowsum,
                           int M, int mPer) {
    const int lane = threadIdx.x;
    const int lo   = lane & 15;
    const int hi   = lane >> 4;            // 0: lanes 0-15, 1: lanes 16-31
    const int rowTile = blockIdx.x * 16;
    const int m0base  = blockIdx.y * mPer;

    // A-matrix 16x4 f32: lanes 0-15 -> K=0,1 (x,y); lanes 16-31 -> K=2,3 (z,0)
    const int qrow = rowTile + lo;
    const float ax = q[qrow * 3 + 0];
    const float ay = q[qrow * 3 + 1];
    const float az = q[qrow * 3 + 2];
    v2f A;
    A.x = hi ? az : ax;
    A.y = hi ? 0.0f : ay;

    // |q|^2 for the 8 D rows this lane owns (VGPR r -> row r + hi*8)
    float q2v[8];
#pragma unroll
    for (int r = 0; r < 8; ++r) {
        const int rr = rowTile + r + hi * 8;
        const float x = q[rr * 3 + 0], y = q[rr * 3 + 1], z = q[rr * 3 + 2];
        q2v[r] = x * x + y * y + z * z;
    }

    float acc[8];
#pragma unroll
    for (int r = 0; r < 8; ++r) acc[r] = 0.0f;

    for (int m0 = m0base; m0 < m0base + mPer; m0 += 16) {
        const int col = m0 + lo;
        // every lane loads its column's full coords: no cross-lane traffic
        const float cx = c[col * 3 + 0];
        const float cy = c[col * 3 + 1];
        const float cz = c[col * 3 + 2];
        // B-matrix 4x16 f32: column = lo; lanes 0-15 -> K=0,1; lanes 16-31 -> K=2,3
        v2f B;
        B.x = hi ? cz : cx;
        B.y = hi ? 0.0f : cy;
        const float c2 = cx * cx + cy * cy + cz * cz;

        v8f D = {};
        D = __builtin_amdgcn_wmma_f32_16x16x4_f32(
                false, A, false, B, (short)0, D, false, false);

#pragma unroll
        for (int r = 0; r < 8; ++r) {
            const float d2 = q2v[r] + c2 - 2.0f * D[r];
            acc[r] += fast_w(d2);   // sigma=1 -> exp(-dist/2)
        }
    }

    // reduce over 16 lanes within each half (once per block)
#pragma unroll
    for (int off = 1; off < 16; off <<= 1) {
#pragma unroll
        for (int r = 0; r < 8; ++r) acc[r] += __shfl_xor(acc[r], off);
    }
    if (lo == 0) {
#pragma unroll
        for (int r = 0; r < 8; ++r)
            atomicAdd(&rowsum[rowTile + r + hi * 8], acc[r]);
    }
}

// ---------------- pass 2: normalized weights + fused feature GEMV ----------------
__global__ void svg_weights(const float* __restrict__ q,
                            const float* __restrict__ c,
                            const float* __restrict__ feat,   // [M][4]
                            const float* __restrict__ vs,     // [M]
                            const float* __restrict__ rowsum, // [N]
                            float* __restrict__ wout,         // [N][M]
                            float* __restrict__ facc,         // [N][5]
                            int M, int mPer) {
    const int lane = threadIdx.x;
    const int lo   = lane & 15;
    const int hi   = lane >> 4;
    const int rowTile = blockIdx.x * 16;
    const int m0base  = blockIdx.y * mPer;

    const int qrow = rowTile + lo;
    const float ax = q[qrow * 3 + 0];
    const float ay = q[qrow * 3 + 1];
    const float az = q[qrow * 3 + 2];
    v2f A;
    A.x = hi ? az : ax;
    A.y = hi ? 0.0f : ay;

    float q2v[8], inv[8];
    float* wptr[8];
#pragma unroll
    for (int r = 0; r < 8; ++r) {
        const int rr = rowTile + r + hi * 8;
        const float x = q[rr * 3 + 0], y = q[rr * 3 + 1], z = q[rr * 3 + 2];
        q2v[r] = x * x + y * y + z * z;
        inv[r] = 1.0f / (rowsum[rr] + EPS_F);
        wptr[r] = wout + (size_t)rr * (size_t)M;
    }

    float fa[8][5];
#pragma unroll
    for (int r = 0; r < 8; ++r)
#pragma unroll
        for (int j = 0; j < 5; ++j) fa[r][j] = 0.0f;

    for (int m0 = m0base; m0 < m0base + mPer; m0 += 16) {
        const int col = m0 + lo;
        const float cx = c[col * 3 + 0];
        const float cy = c[col * 3 + 1];
        const float cz = c[col * 3 + 2];
        v2f B;
        B.x = hi ? cz : cx;
        B.y = hi ? 0.0f : cy;
        const float c2 = cx * cx + cy * cy + cz * cz;

        v8f D = {};
        D = __builtin_amdgcn_wmma_f32_16x16x4_f32(
                false, A, false, B, (short)0, D, false, false);

        const float f0 = feat[col * 4 + 0];
        const float f1 = feat[col * 4 + 1];
        const float f2 = feat[col * 4 + 2];
        const float f3 = feat[col * 4 + 3];
        const float sz = vs[col];

#pragma unroll
        for (int r = 0; r < 8; ++r) {
            const float d2 = q2v[r] + c2 - 2.0f * D[r];
            const float w  = fast_w(d2) * inv[r];
            wptr[r][col] = w;
            fa[r][0] += w * f0;
            fa[r][1] += w * f1;
            fa[r][2] += w * f2;
            fa[r][3] += w * f3;
            fa[r][4] += w * sz;
        }
    }

#pragma unroll
    for (int off = 1; off < 16; off <<= 1) {
#pragma unroll
        for (int r = 0; r < 8; ++r)
#pragma unroll
            for (int j = 0; j < 5; ++j) fa[r][j] += __shfl_xor(fa[r][j], off);
    }
    if (lo == 0) {
#pragma unroll
        for (int r = 0; r < 8; ++r) {
            const int rr = rowTile + r + hi * 8;
#pragma unroll
            for (int j = 0; j < 5; ++j)
                atomicAdd(&facc[rr * 5 + j], fa[r][j]);
        }
    }
}

// ---------------- pass 3: finalize (softplus / sigmoid / sizes) ----------------
__global__ void svg_finalize(const float* __restrict__ facc,  // [N][5]
                             float* __restrict__ out, int N) {
    int n = blockIdx.x * blockDim.x + threadIdx.x;
    if (n >= N) return;
    const float f0 = facc[n * 5 + 0];
    const float f1 = facc[n * 5 + 1];
    const float f2 = facc[n * 5 + 2];
    const float f3 = facc[n * 5 + 3];
    const float f4 = facc[n * 5 + 4];
    // softplus(x) = max(x,0) + log1p(exp(-|x|))
    out[n] = fmaxf(f0, 0.0f) + log1pf(__expf(-fabsf(f0)));
    out[N + n * 3 + 0] = 1.0f / (1.0f + __expf(-f1));
    out[N + n * 3 + 1] = 1.0f / (1.0f + __expf(-f2));
    out[N + n * 3 + 2] = 1.0f / (1.0f + __expf(-f3));
    out[4 * N + n] = f4;
}

extern "C" void kernel_launch(void* const* d_in, const int* in_sizes, int n_in,
                              void* d_out, int out_size, void* d_ws, size_t ws_size,
                              hipStream_t stream) {
    const float* q    = (const float*)d_in[0];  // [N][3]
    const float* c    = (const float*)d_in[1];  // [M][3]
    const float* feat = (const float*)d_in[2];  // [M][4]
    const float* vs   = (const float*)d_in[3];  // [M]
    float* out = (float*)d_out;

    const int N = in_sizes[0] / 3;   // 16384
    const int M = in_sizes[1] / 3;   // 8192

    float* rowsum = (float*)d_ws;          // [N]
    float* facc   = rowsum + N;            // [N][5]

    const int nChunks = 8;
    const int mPer = M / nChunks;          // 1024, multiple of 16

    // zero rowsum + facc (ws is poisoned by harness; must re-zero every call)
    {
        int total = N * 6;
        svg_zero_ws<<<(total + 255) / 256, 256, 0, stream>>>((float*)d_ws, total);
    }

    dim3 grid(N / 16, nChunks);
    svg_rowsum<<<grid, 32, 0, stream>>>(q, c, rowsum, M, mPer);

    float* wout = out + (size_t)5 * N;     // weights block of d_out
    svg_weights<<<grid, 32, 0, stream>>>(q, c, feat, vs, rowsum, wout, facc, M, mPer);

    svg_finalize<<<(N + 255) / 256, 256, 0, stream>>>(facc, out, N);
}